// Gaussian3D_73040213836087
// MI455X (gfx1250) — compile-verified
//
#include <hip/hip_runtime.h>
#include <hip/hip_bf16.h>

typedef float v4f __attribute__((ext_vector_type(4)));

#define TILE 256

// 3D Gaussian covariance: cov[n,i,k] = sum_j R[n,i,j] * exp(ls[n,j]) * R[n,k,j]
__global__ __launch_bounds__(TILE) void Gaussian3D_cov_kernel(
    const float* __restrict__ quat,    // [N,4] (w,x,y,z)
    const float* __restrict__ lscale,  // [N,3]
    float* __restrict__ out,           // [N,3,3]
    int n)
{
    __shared__ float lsStage[3 * TILE];   // coalesced-staged log_scale
    __shared__ float covStage[9 * TILE];  // store staging for coalesced b128 drain

    const int tid       = threadIdx.x;
    const int blockBase = (int)blockIdx.x * TILE;
    const int p         = blockBase + tid;
    const bool fullBlock = (blockBase + TILE) <= n;

    // ---- Stage log_scale: 3 coalesced b32 streams via CDNA5 async global->LDS ----
    if (fullBlock) {
        // Low 32 bits of a flat pointer into the LDS aperture ARE the wave-relative
        // LDS byte address (ISA 10.2).
        unsigned lds_a = (unsigned)(size_t)&lsStage[tid];
        unsigned goff  = (unsigned)(((size_t)blockBase * 3 + (size_t)tid) * sizeof(float));
        // GVS form: vdst(LDS addr), vaddr(32-bit offset), saddr(64-bit base).
        // INST_OFFSET is added to BOTH the global and LDS addresses, which matches
        // the lsStage[tid + j*256] <- ls[base + tid + j*256] layout exactly.
        asm volatile(
            "global_load_async_to_lds_b32 %0, %1, %2\n\t"
            "global_load_async_to_lds_b32 %0, %1, %2 offset:1024\n\t"
            "global_load_async_to_lds_b32 %0, %1, %2 offset:2048"
            :
            : "v"(lds_a), "v"(goff), "s"(lscale)
            : "memory");
    } else {
        const int m = n - blockBase;  // 0 < m < TILE
        for (int k = tid; k < 3 * m; k += TILE)
            lsStage[k] = lscale[(size_t)blockBase * 3 + k];
    }

    // ---- Quaternion: streaming (non-temporal) b128 load, naturally coalesced ----
    v4f q = {1.0f, 0.0f, 0.0f, 0.0f};
    if (p < n)
        q = __builtin_nontemporal_load((const v4f*)quat + p);

    asm volatile("s_wait_asynccnt 0" ::: "memory");
    __syncthreads();

    const float lx = lsStage[3 * tid + 0];
    const float ly = lsStage[3 * tid + 1];
    const float lz = lsStage[3 * tid + 2];

    // ---- Normalize quaternion (reference re-normalizes) ----
    float qw = q.x, qx = q.y, qy = q.z, qz = q.w;
    const float nrm = qw * qw + qx * qx + qy * qy + qz * qz;
    const float inv = 1.0f / sqrtf(nrm);
    qw *= inv; qx *= inv; qy *= inv; qz *= inv;

    const float xx = qx * qx, yy = qy * qy, zz = qz * qz;
    const float xy = qx * qy, xz = qx * qz, yz = qy * qz;
    const float wx = qw * qx, wy = qw * qy, wz = qw * qz;

    const float r00 = 1.0f - 2.0f * (yy + zz);
    const float r01 = 2.0f * (xy - wz);
    const float r02 = 2.0f * (xz + wy);
    const float r10 = 2.0f * (xy + wz);
    const float r11 = 1.0f - 2.0f * (xx + zz);
    const float r12 = 2.0f * (yz - wx);
    const float r20 = 2.0f * (xz - wy);
    const float r21 = 2.0f * (yz + wx);
    const float r22 = 1.0f - 2.0f * (xx + yy);

    const float sx = expf(lx);
    const float sy = expf(ly);
    const float sz = expf(lz);

    // cov[i][k] = sx*R[i][0]*R[k][0] + sy*R[i][1]*R[k][1] + sz*R[i][2]*R[k][2]
    const float a0 = sx * r00, a1 = sy * r01, a2 = sz * r02;
    const float b0 = sx * r10, b1 = sy * r11, b2 = sz * r12;
    const float d0 = sx * r20, d1 = sy * r21, d2 = sz * r22;

    const float c00 = a0 * r00 + a1 * r01 + a2 * r02;
    const float c01 = a0 * r10 + a1 * r11 + a2 * r12;
    const float c02 = a0 * r20 + a1 * r21 + a2 * r22;
    const float c11 = b0 * r10 + b1 * r11 + b2 * r12;
    const float c12 = b0 * r20 + b1 * r21 + b2 * r22;
    const float c22 = d0 * r20 + d1 * r21 + d2 * r22;

    // Stage symmetric 3x3 into LDS (stride-9 word writes: gcd(9,64)=1, conflict-free)
    float* w = &covStage[tid * 9];
    w[0] = c00; w[1] = c01; w[2] = c02;
    w[3] = c01; w[4] = c11; w[5] = c12;
    w[6] = c02; w[7] = c12; w[8] = c22;

    __syncthreads();

    // ---- Drain as fully coalesced non-temporal b128 stores (576 float4 / block) ----
    if (fullBlock) {
        const v4f* __restrict__ src = (const v4f*)covStage;
        v4f* __restrict__ dst = (v4f*)(out + (size_t)blockBase * 9);
        for (int idx = tid; idx < (9 * TILE) / 4; idx += TILE)
            __builtin_nontemporal_store(src[idx], dst + idx);
    } else {
        const int m = n - blockBase;
        for (int k = tid; k < 9 * m; k += TILE)
            out[(size_t)blockBase * 9 + k] = covStage[k];
    }
}

extern "C" void kernel_launch(void* const* d_in, const int* in_sizes, int n_in,
                              void* d_out, int out_size, void* d_ws, size_t ws_size,
                              hipStream_t stream) {
    const float* quat   = (const float*)d_in[0];  // [N,4]
    const float* lscale = (const float*)d_in[1];  // [N,3]
    float* out          = (float*)d_out;          // [N,3,3]

    const int n      = in_sizes[0] / 4;
    const int blocks = (n + TILE - 1) / TILE;

    Gaussian3D_cov_kernel<<<blocks, TILE, 0, stream>>>(quat, lscale, out, n);
}